// EdgeLinear_16088947491452
// MI455X (gfx1250) — compile-verified
//
#include <hip/hip_runtime.h>

typedef __attribute__((ext_vector_type(2))) float v2f;
typedef __attribute__((ext_vector_type(4))) float v4f;
typedef __attribute__((ext_vector_type(8))) float v8f;

#define FE   32
#define FN   64
#define FG   64
#define FOUT 64

#define TILE_M 16
#define WAVES_PER_BLOCK 8
#define LDS_STRIDE 68   // floats; pad 16x64 tile to kill bank conflicts

// ---------------------------------------------------------------------------
// gproj[g][o] = dot(global_features[g], W_global[o]) + bias[o]   (16x64, tiny)
// ---------------------------------------------------------------------------
__global__ void global_proj_kernel(const float* __restrict__ gf,
                                   const float* __restrict__ Wg,
                                   const float* __restrict__ bias,
                                   float* __restrict__ gproj, int G) {
    int idx = blockIdx.x * blockDim.x + threadIdx.x;
    if (idx >= G * FOUT) return;
    int g = idx >> 6, o = idx & 63;
    float acc = bias[o];
    #pragma unroll 8
    for (int k = 0; k < FG; ++k) acc += gf[g * FG + k] * Wg[o * FG + k];
    gproj[idx] = acc;
}

// ---------------------------------------------------------------------------
// Node projections via V_WMMA_F32_16X16X4_F32:
//   sproj = node_features @ W_sender^T ; rproj = node_features @ W_receiver^T
// One wave computes a 16-row x 64-col tile for both weights (shared A frag).
// ---------------------------------------------------------------------------
__global__ __launch_bounds__(256) void node_proj_kernel(
    const float* __restrict__ nf,        // [N,64]
    const float* __restrict__ Ws,        // [64,64]
    const float* __restrict__ Wr,        // [64,64]
    float* __restrict__ sproj,           // [N,64]
    float* __restrict__ rproj,           // [N,64]
    int num_tiles) {
    const int lane = threadIdx.x & 31;
    const int wave = (int)((blockIdx.x * blockDim.x + threadIdx.x) >> 5);
    const int nwaves = (int)((gridDim.x * blockDim.x) >> 5);
    const int half = lane >> 4;          // 0: K=kb..kb+1, 1: K=kb+2..kb+3
    const int r    = lane & 15;
    const int koff = half * 2;

    for (int tile = wave; tile < num_tiles; tile += nwaves) {
        const int row0 = tile * TILE_M;
        const float* A = nf + (size_t)row0 * FN;
        v8f accS[4] = {}, accR[4] = {};
        #pragma unroll
        for (int ks = 0; ks < FN / 4; ++ks) {
            const int kb = ks * 4 + koff;
            v2f a = *(const v2f*)(A + r * FN + kb);
            #pragma unroll
            for (int t = 0; t < 4; ++t) {
                v2f bs = *(const v2f*)(Ws + (t * 16 + r) * FN + kb);
                v2f br = *(const v2f*)(Wr + (t * 16 + r) * FN + kb);
                accS[t] = __builtin_amdgcn_wmma_f32_16x16x4_f32(
                    false, a, false, bs, (short)0, accS[t], false, false);
                accR[t] = __builtin_amdgcn_wmma_f32_16x16x4_f32(
                    false, a, false, br, (short)0, accR[t], false, false);
            }
        }
        // C/D layout: vgpr reg, lane -> (M = reg + half*8, N = 16t + r)
        #pragma unroll
        for (int t = 0; t < 4; ++t) {
            #pragma unroll
            for (int reg = 0; reg < 8; ++reg) {
                const int m = reg + half * 8;
                const size_t o = (size_t)(row0 + m) * FOUT + t * 16 + r;
                sproj[o] = accS[t][reg];
                rproj[o] = accR[t][reg];
            }
        }
    }
}

__global__ void node_proj_tail_kernel(const float* __restrict__ nf,
                                      const float* __restrict__ Ws,
                                      const float* __restrict__ Wr,
                                      float* __restrict__ sproj,
                                      float* __restrict__ rproj,
                                      int row_start, int nrows) {
    int idx = blockIdx.x * blockDim.x + threadIdx.x;
    if (idx >= nrows * FOUT) return;
    int m = row_start + idx / FOUT, o = idx % FOUT;
    float as = 0.f, ar = 0.f;
    for (int k = 0; k < FN; ++k) {
        float x = nf[(size_t)m * FN + k];
        as += x * Ws[o * FN + k];
        ar += x * Wr[o * FN + k];
    }
    sproj[(size_t)m * FOUT + o] = as;
    rproj[(size_t)m * FOUT + o] = ar;
}

// ---------------------------------------------------------------------------
// Main edge kernel: per wave, one 16-edge x 64-out tile.
//   WMMA (K=32) -> LDS -> per-lane float4 readback + 3 gathered adds -> store
// ---------------------------------------------------------------------------
__global__ __launch_bounds__(256) void edge_update_kernel(
    const float* __restrict__ ef,        // [E,32]
    const float* __restrict__ We,        // [64,32]
    const float* __restrict__ sproj,     // [N,64]
    const float* __restrict__ rproj,     // [N,64]
    const float* __restrict__ gproj,     // [G,64] (bias folded in)
    const int*   __restrict__ senders,
    const int*   __restrict__ receivers,
    const int*   __restrict__ gids,
    float* __restrict__ out,             // [E,64]
    int num_tiles) {
    __shared__ float lds[WAVES_PER_BLOCK * TILE_M * LDS_STRIDE];
    const int lane = threadIdx.x & 31;
    const int wib  = threadIdx.x >> 5;
    float* wlds = lds + wib * (TILE_M * LDS_STRIDE);
    const int wave   = (int)blockIdx.x * WAVES_PER_BLOCK + wib;
    const int nwaves = (int)gridDim.x * WAVES_PER_BLOCK;

    const int half = lane >> 4;
    const int r    = lane & 15;
    const int koff = half * 2;

    for (int tile = wave; tile < num_tiles; tile += nwaves) {
        const int e0 = tile * TILE_M;
        const float* A = ef + (size_t)e0 * FE;

        v8f acc[4] = {};
        #pragma unroll
        for (int ks = 0; ks < FE / 4; ++ks) {
            const int kb = ks * 4 + koff;
            v2f a = *(const v2f*)(A + r * FE + kb);
            #pragma unroll
            for (int t = 0; t < 4; ++t) {
                v2f b = *(const v2f*)(We + (t * 16 + r) * FE + kb);
                acc[t] = __builtin_amdgcn_wmma_f32_16x16x4_f32(
                    false, a, false, b, (short)0, acc[t], false, false);
            }
        }

        // Spill 16x64 accumulator tile to padded LDS (conflict-free).
        #pragma unroll
        for (int t = 0; t < 4; ++t) {
            #pragma unroll
            for (int reg = 0; reg < 8; ++reg) {
                const int m = reg + half * 8;
                wlds[m * LDS_STRIDE + (t * 16 + r)] = acc[t][reg];
            }
        }
        // Per-wave private LDS region; DS ops are in-order within a wave and
        // the compiler inserts the s_wait_dscnt before readback.

        // Lane handles edge row m = r, column half [half*32, half*32+32).
        const int e  = e0 + r;
        const int s  = senders[e];
        const int rc = receivers[e];
        const int g  = gids[e];
        const float* sp = sproj + (size_t)s  * FOUT + half * 32;  // L2 hit
        const float* rp = rproj + (size_t)rc * FOUT + half * 32;  // L2 hit
        const float* gp = gproj + (size_t)g  * FOUT + half * 32;  // L0/L2 hit
        const float* cl = wlds + r * LDS_STRIDE + half * 32;
        float*       op = out + (size_t)e * FOUT + half * 32;
        #pragma unroll
        for (int j = 0; j < 8; ++j) {
            v4f c  = *(const v4f*)(cl + j * 4);
            v4f vs = *(const v4f*)(sp + j * 4);
            v4f vr = *(const v4f*)(rp + j * 4);
            v4f vg = *(const v4f*)(gp + j * 4);
            *(v4f*)(op + j * 4) = c + vs + vr + vg;
        }
    }
}

__global__ void edge_tail_kernel(const float* __restrict__ ef,
                                 const float* __restrict__ We,
                                 const float* __restrict__ sproj,
                                 const float* __restrict__ rproj,
                                 const float* __restrict__ gproj,
                                 const int* __restrict__ senders,
                                 const int* __restrict__ receivers,
                                 const int* __restrict__ gids,
                                 float* __restrict__ out,
                                 int e_start, int nrem) {
    int idx = blockIdx.x * blockDim.x + threadIdx.x;
    if (idx >= nrem * FOUT) return;
    int e = e_start + idx / FOUT, o = idx % FOUT;
    float a = 0.f;
    for (int k = 0; k < FE; ++k) a += ef[(size_t)e * FE + k] * We[o * FE + k];
    a += sproj[(size_t)senders[e] * FOUT + o];
    a += rproj[(size_t)receivers[e] * FOUT + o];
    a += gproj[(size_t)gids[e] * FOUT + o];
    out[(size_t)e * FOUT + o] = a;
}

// ---------------------------------------------------------------------------
extern "C" void kernel_launch(void* const* d_in, const int* in_sizes, int n_in,
                              void* d_out, int out_size, void* d_ws, size_t ws_size,
                              hipStream_t stream) {
    const float* edge_features = (const float*)d_in[0];
    const float* node_features = (const float*)d_in[1];
    const float* global_feats  = (const float*)d_in[2];
    const float* W_edge        = (const float*)d_in[3];
    const float* W_sender      = (const float*)d_in[4];
    const float* W_receiver    = (const float*)d_in[5];
    const float* W_global      = (const float*)d_in[6];
    const float* bias          = (const float*)d_in[7];
    const int*   senders       = (const int*)d_in[8];
    const int*   receivers     = (const int*)d_in[9];
    const int*   gids          = (const int*)d_in[10];
    float* out = (float*)d_out;

    const int E = in_sizes[8];
    const int N = in_sizes[1] / FN;
    const int G = in_sizes[2] / FG;

    // Workspace: sproj [N,64] | rproj [N,64] | gproj [G,64]
    float* sproj = (float*)d_ws;
    float* rproj = sproj + (size_t)N * FOUT;
    float* gproj = rproj + (size_t)N * FOUT;

    // 1) Global projection + bias (tiny)
    {
        int total = G * FOUT;
        global_proj_kernel<<<(total + 255) / 256, 256, 0, stream>>>(
            global_feats, W_global, bias, gproj, G);
    }

    // 2) Node projections (WMMA), both weights per tile
    {
        int tiles = N / TILE_M;
        if (tiles > 0) {
            int blocks = (tiles + WAVES_PER_BLOCK - 1) / WAVES_PER_BLOCK;
            node_proj_kernel<<<blocks, 256, 0, stream>>>(
                node_features, W_sender, W_receiver, sproj, rproj, tiles);
        }
        int rem = N - tiles * TILE_M;
        if (rem > 0) {
            int total = rem * FOUT;
            node_proj_tail_kernel<<<(total + 255) / 256, 256, 0, stream>>>(
                node_features, W_sender, W_receiver, sproj, rproj,
                tiles * TILE_M, rem);
        }
    }

    // 3) Main edge kernel (WMMA + gathers), memory-bound: ~640MB HBM traffic
    {
        int tiles = E / TILE_M;
        if (tiles > 0) {
            int blocks = 4096;  // grid-stride: ~3 tiles/wave, amortize W loads
            int max_blocks = (tiles + WAVES_PER_BLOCK - 1) / WAVES_PER_BLOCK;
            if (blocks > max_blocks) blocks = max_blocks;
            edge_update_kernel<<<blocks, 256, 0, stream>>>(
                edge_features, W_edge, sproj, rproj, gproj,
                senders, receivers, gids, out, tiles);
        }
        int rem = E - tiles * TILE_M;
        if (rem > 0) {
            int total = rem * FOUT;
            edge_tail_kernel<<<(total + 255) / 256, 256, 0, stream>>>(
                edge_features, W_edge, sproj, rproj, gproj,
                senders, receivers, gids, out, tiles * TILE_M, rem);
        }
    }
}